// TtMambaSSM_15195594293828
// MI455X (gfx1250) — compile-verified
//
#include <hip/hip_runtime.h>
#include <hip/hip_bf16.h>

// Problem dimensions (fixed by the reference).
#define U_ 128
#define H_ 5120
#define R_ 160
#define N_ 32

typedef float v2f __attribute__((ext_vector_type(2)));
typedef float v8f __attribute__((ext_vector_type(8)));

// ---------------------------------------------------------------------------
// Generic fp32 WMMA GEMM: C(M x N) = A(M x K) * B(K x N), all row-major.
// One wave32 per 16x16 output tile, K-loop in steps of 4 using
// V_WMMA_F32_16X16X4_F32.  M % 16 == 0, N % 16 == 0, K % 4 == 0.
// ---------------------------------------------------------------------------
__global__ void wmma_gemm_f32(const float* __restrict__ A,
                              const float* __restrict__ B,
                              float* __restrict__ C,
                              int M, int N, int K) {
    const int lane = threadIdx.x & 31;
    const int wave = threadIdx.x >> 5;
    const int ntn  = N >> 4;                       // tiles along N
    const int tile = blockIdx.x * (blockDim.x >> 5) + wave;
    if (tile >= (M >> 4) * ntn) return;            // wave-uniform: EXEC stays all-1s
    const int tm = tile / ntn;
    const int tn = tile - tm * ntn;

    // A fragment (16x4 f32): lane l -> row m = l&15, k-halves 2*(l>>4)+{0,1}
    // B fragment (4x16 f32): lane l -> col n = l&15, same k-halves
    const int mr = lane & 15;
    const int kh = (lane >> 4) << 1;
    const float* ap = A + (size_t)(tm * 16 + mr) * K + kh;
    const float* bp = B + (size_t)kh * N + (tn * 16 + mr);

    v8f acc = {};
    for (int k0 = 0; k0 < K; k0 += 4) {
        v2f a; a.x = ap[0]; a.y = ap[1];
        v2f b; b.x = bp[0]; b.y = bp[N];
        ap += 4;
        bp += (size_t)4 * N;
        // 8 args: (neg_a, A, neg_b, B, c_mod, C, reuse_a, reuse_b)
        acc = __builtin_amdgcn_wmma_f32_16x16x4_f32(false, a, false, b,
                                                    (short)0, acc, false, false);
    }

    // C/D 16x16 f32 layout: VGPR v, lanes 0-15 -> M=v, lanes 16-31 -> M=v+8
    float* cp = C + (size_t)(tm * 16 + (lane >> 4) * 8) * N + (tn * 16 + mr);
#pragma unroll
    for (int v = 0; v < 8; ++v) cp[(size_t)v * N] = acc[v];
}

// ---------------------------------------------------------------------------
// Fused SSM recurrence (memory-bound: streams h_prev once, 84 MB).
//   delta = softplus(z + b_dt)   (thresh 20)
//   h_new = exp(delta*A)*h_prev + delta*B*x
//   out   = x*D + sum_n h_new*C
// One wave handles 4 h-rows: each lane loads a float4 (4 states), so a wave
// reads 128 contiguous floats of h_prev (fully coalesced B128 per lane).
// The N-contraction reduces over the 8 lanes sharing an h-row via shfl_xor.
// grid = (H/32, U), block = 256 (8 waves).
// ---------------------------------------------------------------------------
__global__ void mamba_ssm_fused(const float* __restrict__ x,
                                const float* __restrict__ h_prev,
                                const float* __restrict__ z,
                                const float* __restrict__ b_dt,
                                const float* __restrict__ Bm,
                                const float* __restrict__ Cm,
                                const float* __restrict__ Amat,
                                const float* __restrict__ Dv,
                                float* __restrict__ out) {
    const int lane = threadIdx.x & 31;
    const int wave = threadIdx.x >> 5;
    const int u    = blockIdx.y;
    const int h0   = (blockIdx.x * 8 + wave) * 4;      // first of 4 h-rows
    const int h    = h0 + (lane >> 3);                  // this lane's h-row

    // 128-byte aligned float4 loads (offsets are multiples of 32 floats).
    const float4 hp = ((const float4*)(h_prev + ((size_t)u * H_ + h0) * N_))[lane];
    const float4 av = ((const float4*)(Amat + (size_t)h0 * N_))[lane];
    const float4 bv = ((const float4*)(Bm + (size_t)u * N_))[lane & 7];
    const float4 cv = ((const float4*)(Cm + (size_t)u * N_))[lane & 7];

    const float zb    = z[(size_t)u * H_ + h] + b_dt[h];
    const float delta = (zb > 20.0f) ? zb : logf(1.0f + __expf(zb));
    const float xv    = x[(size_t)u * H_ + h];
    const float dx    = delta * xv;

    float acc = (__expf(delta * av.x) * hp.x + dx * bv.x) * cv.x;
    acc      += (__expf(delta * av.y) * hp.y + dx * bv.y) * cv.y;
    acc      += (__expf(delta * av.z) * hp.z + dx * bv.z) * cv.z;
    acc      += (__expf(delta * av.w) * hp.w + dx * bv.w) * cv.w;

    // Reduce over the 8 lanes (masks 1,2,4 stay inside each group of 8).
    acc += __shfl_xor(acc, 1);
    acc += __shfl_xor(acc, 2);
    acc += __shfl_xor(acc, 4);

    if ((lane & 7) == 0)
        out[(size_t)u * H_ + h] = xv * Dv[h] + acc;
}

// ---------------------------------------------------------------------------
// Host side.
// ---------------------------------------------------------------------------
extern "C" void kernel_launch(void* const* d_in, const int* in_sizes, int n_in,
                              void* d_out, int out_size, void* d_ws, size_t ws_size,
                              hipStream_t stream) {
    (void)in_sizes; (void)n_in; (void)out_size; (void)ws_size;

    const float* x        = (const float*)d_in[0];   // (U,H)
    const float* h_prev   = (const float*)d_in[1];   // (U,H,N)
    const float* W_dt_low = (const float*)d_in[2];   // (H,R)
    const float* W_dt     = (const float*)d_in[3];   // (R,H)
    const float* b_dt     = (const float*)d_in[4];   // (H)
    const float* W_B      = (const float*)d_in[5];   // (H,N)
    const float* W_C      = (const float*)d_in[6];   // (H,N)
    const float* Amat     = (const float*)d_in[7];   // (H,N)
    const float* Dv       = (const float*)d_in[8];   // (H)
    float* out            = (float*)d_out;           // (U,H)

    // Workspace partition (fp32): t (U*R), z (U*H), B (U*N), C (U*N) ~ 2.7 MB.
    float* t  = (float*)d_ws;
    float* zz = t  + (size_t)U_ * R_;
    float* Bm = zz + (size_t)U_ * H_;
    float* Cm = Bm + (size_t)U_ * N_;

    const int BLK = 256;               // 8 waves -> 8 tiles per block
    auto blocks_for = [](int M, int N) { return ((M >> 4) * (N >> 4) + 7) / 8; };

    // t = x @ W_dt_low            (128 x 160, K = 5120)
    wmma_gemm_f32<<<blocks_for(U_, R_), BLK, 0, stream>>>(x, W_dt_low, t, U_, R_, H_);
    // z = t @ W_dt                (128 x 5120, K = 160)   [bias+softplus fused later]
    wmma_gemm_f32<<<blocks_for(U_, H_), BLK, 0, stream>>>(t, W_dt, zz, U_, H_, R_);
    // B = x @ W_B, C = x @ W_C    (128 x 32, K = 5120)
    wmma_gemm_f32<<<blocks_for(U_, N_), BLK, 0, stream>>>(x, W_B, Bm, U_, N_, H_);
    wmma_gemm_f32<<<blocks_for(U_, N_), BLK, 0, stream>>>(x, W_C, Cm, U_, N_, H_);

    // Fused recurrence + output projection (streams h_prev once).
    dim3 grid(H_ / 32, U_);
    mamba_ssm_fused<<<grid, BLK, 0, stream>>>(x, h_prev, zz, b_dt, Bm, Cm, Amat, Dv, out);
}